// CONV_GNN_48722109005962
// MI455X (gfx1250) — compile-verified
//
#include <hip/hip_runtime.h>
#include <hip/hip_bf16.h>

typedef __attribute__((ext_vector_type(2))) float v2f;
typedef __attribute__((ext_vector_type(8))) float v8f;

#define N_NODES 100000
#define ROW_TILES (N_NODES / 16)   // 6250 exactly

// ---------------------------------------------------------------- degree prep
__global__ void deg_init_kernel(float* deg, int n) {
    int i = blockIdx.x * blockDim.x + threadIdx.x;
    if (i < n) deg[i] = 1.0f;  // self-loop contributes 1 to every node's degree
}

__global__ void deg_count_kernel(const int* __restrict__ col, float* deg, int E) {
    int e = blockIdx.x * blockDim.x + threadIdx.x;
    if (e < E) unsafeAtomicAdd(&deg[col[e]], 1.0f);
}

__global__ void deg_rsqrt_kernel(float* deg, int n) {
    int i = blockIdx.x * blockDim.x + threadIdx.x;
    if (i < n) deg[i] = rsqrtf(deg[i]);  // deg >= 1 always
}

// ---------------------------------------------------------------- WMMA GEMM 1
// h1[N,64] = x[N,128] @ W1[128,64].  8 waves/block: 2 row-tiles x 4 col-tiles.
__global__ __launch_bounds__(256) void gemm1_kernel(const float* __restrict__ x,
                                                    const float* __restrict__ W,
                                                    float* __restrict__ h) {
    const int lane = threadIdx.x & 31;
    const int wave = threadIdx.x >> 5;
    const int rowTile = blockIdx.x * 2 + (wave >> 2);
    const int colTile = wave & 3;
    const int m0 = rowTile * 16;
    const int n0 = colTile * 16;
    const int half = lane >> 4;   // 0: K pair {k0,k0+1}, 1: {k0+2,k0+3}
    const int l    = lane & 15;

    v8f c = {};
    const float* arow = x + (m0 + l) * 128 + 2 * half;      // M=l, K base
    const float* bcol = W + (2 * half) * 64 + n0 + l;       // N=l, K base
#pragma unroll
    for (int k0 = 0; k0 < 128; k0 += 4) {
        v2f a = *(const v2f*)(arow + k0);                   // aligned float2
        v2f b;
        b.x = bcol[(k0 + 0) * 64];
        b.y = bcol[(k0 + 1) * 64];
        c = __builtin_amdgcn_wmma_f32_16x16x4_f32(false, a, false, b,
                                                  (short)0, c, false, false);
    }
    float* o = h + (m0 + half * 8) * 64 + n0 + l;           // C/D layout store
#pragma unroll
    for (int v = 0; v < 8; ++v) o[v * 64] = c[v];
}

// ---------------------------------------------------------------- WMMA GEMM 2
// g[N,32] = a[N,64] @ W2[64,32].  8 waves/block: 4 row-tiles x 2 col-tiles.
__global__ __launch_bounds__(256) void gemm2_kernel(const float* __restrict__ ain,
                                                    const float* __restrict__ W,
                                                    float* __restrict__ g) {
    const int lane = threadIdx.x & 31;
    const int wave = threadIdx.x >> 5;
    const int rowTile = blockIdx.x * 4 + (wave >> 1);
    if (rowTile >= ROW_TILES) return;                       // whole wave exits
    const int colTile = wave & 1;
    const int m0 = rowTile * 16;
    const int n0 = colTile * 16;
    const int half = lane >> 4;
    const int l    = lane & 15;

    v8f c = {};
    const float* arow = ain + (m0 + l) * 64 + 2 * half;
    const float* bcol = W + (2 * half) * 32 + n0 + l;
#pragma unroll
    for (int k0 = 0; k0 < 64; k0 += 4) {
        v2f a = *(const v2f*)(arow + k0);
        v2f b;
        b.x = bcol[(k0 + 0) * 32];
        b.y = bcol[(k0 + 1) * 32];
        c = __builtin_amdgcn_wmma_f32_16x16x4_f32(false, a, false, b,
                                                  (short)0, c, false, false);
    }
    float* o = g + (m0 + half * 8) * 32 + n0 + l;
#pragma unroll
    for (int v = 0; v < 8; ++v) o[v * 32] = c[v];
}

// ------------------------------------------------- bias + self-loop init
// acc[i,f] = bias[f] + h[i,f] * dis[i]^2   (self-edge norm = dis*dis)
__global__ void init_acc_kernel(const float* __restrict__ h,
                                const float* __restrict__ bias,
                                const float* __restrict__ dis,
                                float* __restrict__ acc,
                                int fshift, int fmask, int total) {
    int i = blockIdx.x * blockDim.x + threadIdx.x;
    if (i >= total) return;
    int node = i >> fshift;
    int f = i & fmask;
    float d = dis[node];
    acc[i] = bias[f] + h[i] * d * d;
}

// ------------------------------------------------- edge gather+scatter, F=64
// One thread handles one edge x 4 features (float4 gather, 4 fp32 atomics).
__global__ void scatter64_kernel(const int* __restrict__ row,
                                 const int* __restrict__ col,
                                 const float* __restrict__ dis,
                                 const float* __restrict__ h,
                                 float* acc, int E) {
    int tid = blockIdx.x * blockDim.x + threadIdx.x;
    if (tid >= E * 16) return;
    int e  = tid >> 4;
    int fc = (tid & 15) << 2;
    int r = row[e], c = col[e];
    float norm = dis[r] * dis[c];
    float4 v = *(const float4*)(h + r * 64 + fc);
    float* p = acc + c * 64 + fc;
    unsafeAtomicAdd(p + 0, v.x * norm);
    unsafeAtomicAdd(p + 1, v.y * norm);
    unsafeAtomicAdd(p + 2, v.z * norm);
    unsafeAtomicAdd(p + 3, v.w * norm);
}

// ------------------------------------------------- edge gather+scatter, F=32
__global__ void scatter32_kernel(const int* __restrict__ row,
                                 const int* __restrict__ col,
                                 const float* __restrict__ dis,
                                 const float* __restrict__ h,
                                 float* acc, int E) {
    int tid = blockIdx.x * blockDim.x + threadIdx.x;
    if (tid >= E * 8) return;
    int e  = tid >> 3;
    int fc = (tid & 7) << 2;
    int r = row[e], c = col[e];
    float norm = dis[r] * dis[c];
    float4 v = *(const float4*)(h + r * 32 + fc);
    float* p = acc + c * 32 + fc;
    unsafeAtomicAdd(p + 0, v.x * norm);
    unsafeAtomicAdd(p + 1, v.y * norm);
    unsafeAtomicAdd(p + 2, v.z * norm);
    unsafeAtomicAdd(p + 3, v.w * norm);
}

__global__ void relu_kernel(float* a, int total) {
    int i = blockIdx.x * blockDim.x + threadIdx.x;
    if (i < total) a[i] = fmaxf(a[i], 0.0f);
}

// ---------------------------------------------------------------- launch
extern "C" void kernel_launch(void* const* d_in, const int* in_sizes, int n_in,
                              void* d_out, int out_size, void* d_ws, size_t ws_size,
                              hipStream_t stream) {
    const float* x  = (const float*)d_in[0];
    const int*   ei = (const int*)d_in[1];
    const float* W1 = (const float*)d_in[2];
    const float* b1 = (const float*)d_in[3];
    const float* W2 = (const float*)d_in[4];
    const float* b2 = (const float*)d_in[5];
    float* out = (float*)d_out;

    const int E = in_sizes[1] / 2;
    const int* row = ei;        // sources
    const int* col = ei + E;    // targets

    // workspace carve-up (all 16B aligned: N_NODES*4 = 400000 divisible by 16)
    float* dis  = (float*)d_ws;                 // N            (degree -> rsqrt)
    float* h1   = dis  + N_NODES;               // N*64   x@W1
    float* acc1 = h1   + N_NODES * 64;          // N*64   layer-1 accumulator
    float* g2   = acc1 + N_NODES * 64;          // N*32   relu(acc1)@W2

    const int T = 256;
    // degree / normalization
    deg_init_kernel<<<(N_NODES + T - 1) / T, T, 0, stream>>>(dis, N_NODES);
    deg_count_kernel<<<(E + T - 1) / T, T, 0, stream>>>(col, dis, E);
    deg_rsqrt_kernel<<<(N_NODES + T - 1) / T, T, 0, stream>>>(dis, N_NODES);

    // layer 1
    gemm1_kernel<<<ROW_TILES / 2, T, 0, stream>>>(x, W1, h1);
    init_acc_kernel<<<(N_NODES * 64 + T - 1) / T, T, 0, stream>>>(
        h1, b1, dis, acc1, 6, 63, N_NODES * 64);
    scatter64_kernel<<<(E * 16 + T - 1) / T, T, 0, stream>>>(row, col, dis, h1, acc1, E);
    relu_kernel<<<(N_NODES * 64 + T - 1) / T, T, 0, stream>>>(acc1, N_NODES * 64);

    // layer 2
    gemm2_kernel<<<(ROW_TILES + 3) / 4, T, 0, stream>>>(acc1, W2, g2);
    init_acc_kernel<<<(N_NODES * 32 + T - 1) / T, T, 0, stream>>>(
        g2, b2, dis, out, 5, 31, N_NODES * 32);
    scatter32_kernel<<<(E * 8 + T - 1) / T, T, 0, stream>>>(row, col, dis, g2, out, E);
}